// FastMultiheadAttention_87608742904300
// MI455X (gfx1250) — compile-verified
//
#include <hip/hip_runtime.h>
#include <hip/hip_bf16.h>

#define DIM    4096
#define BATCH  8
#define NH     32
#define HD     128
#define MAXSEQ 2048
#define CBLK   16

typedef __attribute__((ext_vector_type(2))) float v2f;
typedef __attribute__((ext_vector_type(8))) float v8f;

// Zero the GEMM accumulation buffers; preload residual into the final output
// (K-split partial sums are atomically added on top).
__global__ void init_kernel(float* __restrict__ qws, float* __restrict__ kws,
                            float* __restrict__ vws, float* __restrict__ attnws,
                            float* __restrict__ out, const float* __restrict__ residual) {
    int i = blockIdx.x * blockDim.x + threadIdx.x;
    if (i < BATCH * DIM) {
        qws[i] = 0.f; kws[i] = 0.f; vws[i] = 0.f; attnws[i] = 0.f;
        out[i] = residual[i];
    }
}

// Skinny GEMM: out[m,n] += sum_k A[m,k] * W[n,k], M=8 (padded to 16), N=K=4096.
// One wave per 16x16 output tile, fp32 WMMA (V_WMMA_F32_16X16X4_F32), K split
// across grid.y, partials combined via global_atomic_add_f32.
//
// A-matrix rows 8..15 are don't-care: we feed duplicates of rows 0..7 (no
// predication -> EXEC stays all-ones, no exec save/restore in the hot loop).
// Two accumulators break the WMMA->WMMA dependency chain.
template<int KSPLIT>
__global__ void gemm_wmma_f32(const float* __restrict__ A,
                              const float* __restrict__ W,
                              float* __restrict__ out) {
    const int n0     = blockIdx.x * 16;
    const int kchunk = DIM / KSPLIT;          // in floats
    const int kbeg   = blockIdx.y * kchunk;
    const int lane   = threadIdx.x;
    const int half   = lane >> 4;             // 0: K={0,1}, 1: K={2,3}
    const int lm     = lane & 15;

    // v2f-unit pointers; per-step element offset (k + 2*half) -> v2f idx k/2+half
    const v2f* __restrict__ ap =
        (const v2f*)(A + (size_t)(lm & (BATCH - 1)) * DIM);
    const v2f* __restrict__ wp =
        (const v2f*)(W + (size_t)(n0 + lm) * DIM);
    int idx = (kbeg >> 1) + half;

    v8f c0 = {}, c1 = {};
    const int nsteps = kchunk / 4;            // WMMA K-steps in this chunk
    #pragma unroll 8
    for (int s = 0; s < nsteps; s += 2) {
        v2f a0 = ap[idx + 2 * s];
        v2f b0 = wp[idx + 2 * s];
        v2f a1 = ap[idx + 2 * s + 2];
        v2f b1 = wp[idx + 2 * s + 2];
        c0 = __builtin_amdgcn_wmma_f32_16x16x4_f32(false, a0, false, b0,
                                                   (short)0, c0, false, false);
        c1 = __builtin_amdgcn_wmma_f32_16x16x4_f32(false, a1, false, b1,
                                                   (short)0, c1, false, false);
    }
    v8f c = c0 + c1;

    // C layout: VGPR r, lanes 0-15 -> C[M=r][N=lm]; only M<8 valid.
    if (half == 0) {
        #pragma unroll
        for (int r = 0; r < 8; ++r)
            unsafeAtomicAdd(&out[(size_t)r * DIM + n0 + lm], c[r]);
    }
}

// RoPE on q (in place) and k, scatter roped-k and v into the output caches.
__global__ void rope_cache_kernel(float* __restrict__ q,
                                  const float* __restrict__ kraw,
                                  const float* __restrict__ v,
                                  const float* __restrict__ freqs,
                                  const int* __restrict__ start_pos,
                                  const int* __restrict__ cont2ctx,
                                  float* __restrict__ cache_k,
                                  float* __restrict__ cache_v) {
    int i = blockIdx.x * blockDim.x + threadIdx.x;     // pair index over B*(DIM/2)
    if (i >= BATCH * (DIM / 2)) return;
    const int b  = i / (DIM / 2);
    const int p  = i % (DIM / 2);
    const int dp = p & (HD / 2 - 1);                   // pair index within head
    const int o  = p * 2;                              // flat dim within 4096

    const float cs = freqs[dp * 2 + 0];
    const float sn = freqs[dp * 2 + 1];

    const size_t qi = (size_t)b * DIM + o;
    float qre = q[qi], qim = q[qi + 1];
    q[qi]     = qre * cs - qim * sn;
    q[qi + 1] = qre * sn + qim * cs;

    float kre = kraw[qi], kim = kraw[qi + 1];
    const int row = cont2ctx[b];
    const int pos = start_pos[0];                      // S == 1
    const size_t base = ((size_t)row * MAXSEQ + pos) * DIM + o;
    cache_k[base]     = kre * cs - kim * sn;
    cache_k[base + 1] = kre * sn + kim * cs;
    cache_v[base]     = v[qi];
    cache_v[base + 1] = v[qi + 1];
}

// One block (128 threads) per (b,h): scores -> softmax -> weighted sum of V.
// Masked positions (t > start_pos) contribute exp(-1e9-max) == 0, so we only
// iterate t in [0, start_pos].
__global__ void attention_kernel(const float* __restrict__ q,
                                 const float* __restrict__ cache_k,
                                 const float* __restrict__ cache_v,
                                 const float* __restrict__ mask,
                                 const int* __restrict__ start_pos,
                                 const int* __restrict__ cont2ctx,
                                 float* __restrict__ attn) {
    const int bh  = blockIdx.x;
    const int b   = bh / NH, h = bh % NH;
    const int tid = threadIdx.x;                       // 0..127

    __shared__ float sq[HD];
    __shared__ float sc[MAXSEQ];
    __shared__ float red[128];

    const int row = cont2ctx[b];
    const int T   = start_pos[0] + 1;                  // S == 1
    sq[tid] = q[(size_t)b * DIM + h * HD + tid];
    __syncthreads();

    const float scale = rsqrtf((float)HD);
    float lmax = -3.0e38f;
    for (int t = tid; t < T; t += 128) {
        const float4* __restrict__ kp = (const float4*)
            (cache_k + ((size_t)row * MAXSEQ + t) * DIM + h * HD);
        float s = 0.f;
        #pragma unroll
        for (int d4 = 0; d4 < HD / 4; ++d4) {
            float4 kv = kp[d4];
            s += sq[d4 * 4 + 0] * kv.x + sq[d4 * 4 + 1] * kv.y +
                 sq[d4 * 4 + 2] * kv.z + sq[d4 * 4 + 3] * kv.w;
        }
        s = s * scale + mask[t];
        sc[t] = s;
        lmax  = fmaxf(lmax, s);
    }
    red[tid] = lmax; __syncthreads();
    for (int off = 64; off > 0; off >>= 1) {
        if (tid < off) red[tid] = fmaxf(red[tid], red[tid + off]);
        __syncthreads();
    }
    const float mx = red[0]; __syncthreads();

    float lsum = 0.f;
    for (int t = tid; t < T; t += 128) {
        float p = __expf(sc[t] - mx);
        sc[t] = p;
        lsum += p;
    }
    red[tid] = lsum; __syncthreads();
    for (int off = 64; off > 0; off >>= 1) {
        if (tid < off) red[tid] += red[tid + off];
        __syncthreads();
    }
    const float inv = 1.f / red[0];

    // attn[d = tid]: coalesced 512B reads of V per t across the block
    float acc = 0.f;
    const float* __restrict__ vp =
        cache_v + (size_t)row * MAXSEQ * DIM + h * HD + tid;
    for (int t = 0; t < T; ++t) acc += sc[t] * vp[(size_t)t * DIM];
    attn[(size_t)b * DIM + h * HD + tid] = acc * inv;
}

extern "C" void kernel_launch(void* const* d_in, const int* in_sizes, int n_in,
                              void* d_out, int out_size, void* d_ws, size_t ws_size,
                              hipStream_t stream) {
    const float* x          = (const float*)d_in[0];
    const float* residual   = (const float*)d_in[1];
    const int*   start_pos  = (const int*)  d_in[2];
    const float* freqs      = (const float*)d_in[3];
    const float* mask       = (const float*)d_in[4];
    const float* cache_k_in = (const float*)d_in[5];
    const float* cache_v_in = (const float*)d_in[6];
    const int*   cont2ctx   = (const int*)  d_in[7];
    const float* wq         = (const float*)d_in[8];
    const float* wk         = (const float*)d_in[9];
    const float* wv         = (const float*)d_in[10];
    const float* wo         = (const float*)d_in[11];

    float* out          = (float*)d_out;
    float* new_cache_k  = out + (size_t)BATCH * DIM;
    float* new_cache_v  = new_cache_k + (size_t)CBLK * MAXSEQ * DIM;

    float* qws    = (float*)d_ws;
    float* kws    = qws  + (size_t)BATCH * DIM;
    float* vws    = kws  + (size_t)BATCH * DIM;
    float* attnws = vws  + (size_t)BATCH * DIM;

    const size_t cache_bytes = (size_t)CBLK * MAXSEQ * DIM * sizeof(float);
    hipMemcpyAsync(new_cache_k, cache_k_in, cache_bytes, hipMemcpyDeviceToDevice, stream);
    hipMemcpyAsync(new_cache_v, cache_v_in, cache_bytes, hipMemcpyDeviceToDevice, stream);

    init_kernel<<<(BATCH * DIM + 255) / 256, 256, 0, stream>>>(
        qws, kws, vws, attnws, out, residual);

    constexpr int KSPLIT = 8;
    dim3 g(DIM / 16, KSPLIT);
    gemm_wmma_f32<KSPLIT><<<g, 32, 0, stream>>>(x, wq, qws);
    gemm_wmma_f32<KSPLIT><<<g, 32, 0, stream>>>(x, wk, kws);
    gemm_wmma_f32<KSPLIT><<<g, 32, 0, stream>>>(x, wv, vws);

    rope_cache_kernel<<<(BATCH * (DIM / 2) + 255) / 256, 256, 0, stream>>>(
        qws, kws, vws, freqs, start_pos, cont2ctx, new_cache_k, new_cache_v);

    attention_kernel<<<BATCH * NH, 128, 0, stream>>>(
        qws, new_cache_k, new_cache_v, mask, start_pos, cont2ctx, attnws);

    gemm_wmma_f32<KSPLIT><<<g, 32, 0, stream>>>(attnws, wo, out);
}